// Loss_90692529422942
// MI455X (gfx1250) — compile-verified
//
#include <hip/hip_runtime.h>
#include <hip/hip_bf16.h>

// CTC forward (alpha recursion), sum reduction, zero_infinity.
// B=32 batches -> 32 workgroups; S=513 states -> 544 threads (17 wave32s).
// Per time step the 4KB log-prob row is staged into LDS with CDNA5
// global_load_async_to_lds_b128 (8-deep pipeline, s_wait_asynccnt), hiding
// HBM latency behind the sequential T=1024 dependency chain.
// Single s_barrier per DP step: async issue for row t+7 happens right after
// the barrier, into the buffer whose readers completed before that barrier.

#define NEGINF (-1e30f)

constexpr int kB = 32;
constexpr int kT = 1024;
constexpr int kV = 1024;
constexpr int kL = 256;
constexpr int kS = 2 * kL + 1;            // 513 extended states
constexpr int kThreads = 544;             // 17 waves of 32 (covers 513 states)
constexpr int kCopyLanes = (kV * 4) / 16; // 256 lanes * 16B = one 4KB row
constexpr int kPipe = 8;                  // async pipeline depth (rows in flight)

__device__ __forceinline__ unsigned ldsAddr(const void* p) {
  // Generic pointers to LDS carry the LDS byte offset in the low 32 bits
  // (flat aperture).
  return (unsigned)(uintptr_t)p;
}

__device__ __forceinline__ void waitAsync6() {
#if __has_builtin(__builtin_amdgcn_s_wait_asynccnt)
  __builtin_amdgcn_s_wait_asynccnt(6);
#else
  asm volatile("s_wait_asynccnt 0x6" ::: "memory");
#endif
}
__device__ __forceinline__ void waitAsync0() {
#if __has_builtin(__builtin_amdgcn_s_wait_asynccnt)
  __builtin_amdgcn_s_wait_asynccnt(0);
#else
  asm volatile("s_wait_asynccnt 0x0" ::: "memory");
#endif
}

__device__ __forceinline__ void asyncRowToLds(unsigned ldsDst, unsigned gvoff,
                                              const float* base) {
  asm volatile("global_load_async_to_lds_b128 %0, %1, %2"
               :: "v"(ldsDst), "v"(gvoff), "s"(base) : "memory");
}

__device__ __forceinline__ float logaddexp3(float a, float b, float c) {
  float m = fmaxf(fmaxf(a, b), c);
  float r = __expf(a - m) + __expf(b - m) + __expf(c - m);
  return m + __logf(r);
}
__device__ __forceinline__ float logaddexp2(float a, float b) {
  float m = fmaxf(a, b);
  return m + __logf(__expf(a - m) + __expf(b - m));
}

__global__ __launch_bounds__(kThreads) void ctc_alpha_kernel(
    const float* __restrict__ lp,       // [B,T,V] log-probs
    const int*   __restrict__ labels,   // [B,L]
    const int*   __restrict__ input_lens,
    const int*   __restrict__ label_lens,
    float*       __restrict__ loss_b)   // [B] per-batch loss
{
  __shared__ float lpbuf[kPipe][kV];    // 32KB: async-staged rows, row r -> buf[r%8]
  __shared__ float alpha[2][kS + 2];    // 2-entry NEG pad at the left

  const int b   = blockIdx.x;
  const int tid = threadIdx.x;
  const int s   = tid;

  const float* lp_b = lp + (size_t)b * kT * kV;
  const int Tin = input_lens[b];
  const int Lb  = label_lens[b];

  // Static per-state data: extended label + skip-transition flag.
  int  ext_s = 0;
  bool skip  = false;
  if (s < kS) {
    if (s & 1) {
      ext_s = labels[b * kL + (s >> 1)];
      if (s >= 3) skip = (labels[b * kL + (s >> 1)] != labels[b * kL + (s >> 1) - 1]);
    }
  }

  // t = 0 init: alpha0[0]=lp[0,blank], alpha0[1]=lp[0,label0], else NEG.
  if (tid < 2) { alpha[0][tid] = NEGINF; alpha[1][tid] = NEGINF; }
  if (s < kS)  { alpha[0][s + 2] = (s < 2) ? lp_b[ext_s] : NEGINF; }

  // Pre-issue async copies of rows 1..kPipe-1 (waves 0..7 only, wave-uniform).
  const unsigned lane16 = (unsigned)tid * 16u;
  if (tid < kCopyLanes) {
    #pragma unroll
    for (int r = 1; r < kPipe; ++r) {
      if (r < kT) {
        asyncRowToLds(ldsAddr(&lpbuf[r & (kPipe - 1)][0]) + lane16,
                      (unsigned)r * (kV * 4u) + lane16, lp_b);
      }
    }
  }

  for (int t = 1; t < kT; ++t) {
    const int rb = t & (kPipe - 1);     // LDS buffer holding row t

    // Issuer waves wait until row t has landed; async loads complete in
    // order, so leaving up to kPipe-2 newer rows in flight is safe.
    if (tid < kCopyLanes) {
      if (t + kPipe - 1 < kT) waitAsync6();
      else                    waitAsync0();
    }
    __syncthreads();  // single barrier per step: row t + alpha(t-1) visible

    // Refill: row t+7 -> buf[(t-1)&7]; that buffer held row t-1, whose
    // readers all completed before the barrier above.
    if (tid < kCopyLanes && (t + kPipe - 1) < kT) {
      asyncRowToLds(ldsAddr(&lpbuf[(t - 1) & (kPipe - 1)][0]) + lane16,
                    (unsigned)(t + kPipe - 1) * (kV * 4u) + lane16, lp_b);
    }

    if (s < kS) {
      const float* ard = alpha[(t & 1) ^ 1];
      float em = lpbuf[rb][ext_s];      // LDS gather
      float a1 = ard[s + 2];
      float a2 = ard[s + 1];
      float a3 = skip ? ard[s] : NEGINF;
      float v  = logaddexp3(a1, a2, a3) + em;
      // Freeze past input length; writes go to the other alpha buffer, and
      // the next iteration's barrier orders them against all readers.
      alpha[t & 1][s + 2] = (t < Tin) ? v : a1;
    }
  }

  __syncthreads();  // publish final alpha to thread 0

  if (tid == 0) {
    const float* af = alpha[(kT - 1) & 1];
    int   sl     = 2 * Lb;
    float a_last = af[sl + 2];
    float a_prev = af[sl + 1];
    float l = -logaddexp2(a_last, a_prev);
    if (l > 1e29f) l = 0.0f;            // zero_infinity=True
    loss_b[b] = l;
  }
}

__global__ void ctc_sum_kernel(const float* __restrict__ loss_b,
                               float* __restrict__ out) {
  if (blockIdx.x == 0 && threadIdx.x == 0) {
    float acc = 0.0f;
    for (int i = 0; i < kB; ++i) acc += loss_b[i];  // fixed order: deterministic
    out[0] = acc;
  }
}

extern "C" void kernel_launch(void* const* d_in, const int* in_sizes, int n_in,
                              void* d_out, int out_size, void* d_ws, size_t ws_size,
                              hipStream_t stream) {
  const float* lp     = (const float*)d_in[0];
  const int*   labels = (const int*)d_in[1];
  const int*   ilen   = (const int*)d_in[2];
  const int*   llen   = (const int*)d_in[3];
  float* loss_b = (float*)d_ws;         // 32 floats of scratch

  ctc_alpha_kernel<<<kB, kThreads, 0, stream>>>(lp, labels, ilen, llen, loss_b);
  ctc_sum_kernel<<<1, 32, 0, stream>>>(loss_b, (float*)d_out);
}